// AttentionLayer_41764261987259
// MI455X (gfx1250) — compile-verified
//
#include <hip/hip_runtime.h>
#include <hip/hip_bf16.h>

// Problem constants (from reference): B=16, T=4096, H=64, D=10, M=H*D=640.
#define B_  16
#define T_  4096
#define H_  64
#define D_  10
#define M_  640
#define TT  16            // t-rows staged per block in the main kernel
#define RSQRT10 0.31622776601683794f

typedef int   v4i __attribute__((ext_vector_type(4)));
typedef float v2f __attribute__((ext_vector_type(2)));
typedef float v8f __attribute__((ext_vector_type(8)));
typedef __attribute__((address_space(1))) v4i as1_v4i;
typedef __attribute__((address_space(3))) v4i as3_v4i;

#define HAS_WMMA_F32X4 __has_builtin(__builtin_amdgcn_wmma_f32_16x16x4_f32)

// ---- CDNA5 async global->LDS copy (16B per lane), tracked by ASYNCcnt ----
__device__ __forceinline__ void async_b128(const void* g, void* l) {
#if __has_builtin(__builtin_amdgcn_global_load_async_to_lds_b128)
  __builtin_amdgcn_global_load_async_to_lds_b128(
      (as1_v4i*)(g), (as3_v4i*)(l), 0, 0);
#else
  unsigned lds = (unsigned)(unsigned long long)(__attribute__((address_space(3))) char*)(l);
  unsigned long long ga = (unsigned long long)(g);
  asm volatile("global_load_async_to_lds_b128 %0, %1, off"
               :: "v"(lds), "v"(ga) : "memory");
#endif
}

__device__ __forceinline__ void wait_async0() {
#if __has_builtin(__builtin_amdgcn_s_wait_asynccnt)
  __builtin_amdgcn_s_wait_asynccnt(0);
#else
  asm volatile("s_wait_asynccnt 0" ::: "memory");
#endif
}

__device__ __forceinline__ float fast_tanh(float x) {
#if __has_builtin(__builtin_amdgcn_tanhf)
  return __builtin_amdgcn_tanhf(x);   // gfx1250 v_tanh_f32
#elif __has_builtin(__builtin_amdgcn_tanh_f32)
  return __builtin_amdgcn_tanh_f32(x);
#else
  return tanhf(x);
#endif
}

// ============================================================================
// Kernel 1: k_sum[b,m] = (1/sqrt(10)) * sum_t key[b,t,m]
// This IS a matmul: ones(1xT) x K(TxM). Implemented with
// V_WMMA_F32_16X16X4_F32: C += ones(16x4) x Ktile(4x16). With A==1 the result
// is invariant to B row permutation, and every lane of D's first VGPR holds
// colsum(N = lane&15), so extraction is trivial and layout-robust.
// Grid (M/16, B) = 640 blocks, 8 waves each; wave reduces a 512-row t-slice
// with two interleaved accumulators (breaks the WMMA C-chain RAW).
// ============================================================================
__global__ __launch_bounds__(256) void ksum_kernel(const float* __restrict__ key,
                                                   float* __restrict__ ksum) {
  __shared__ float sp[8 * 32];
  const int tid   = threadIdx.x;
  const int lane  = tid & 31;
  const int w     = tid >> 5;          // 8 waves -> 8 t-slices of 512 rows
  const int m0    = blockIdx.x * 16;
  const int b     = blockIdx.y;
  const int col   = lane & 15;
  const int rhalf = lane >> 4;         // which of the two rows this half-wave holds

  const size_t base = ((size_t)b * T_ + (size_t)w * (T_ / 8)) * M_ + (size_t)(m0 + col);
  const float* p0 = key + base + (size_t)rhalf * M_;        // tile rows {0,1}
  const float* p1 = key + base + (size_t)(rhalf + 2) * M_;  // tile rows {2,3}

  v8f c0 = {};
  v8f c1 = {};
#if HAS_WMMA_F32X4
  const v2f ones = {1.f, 1.f};
#endif

  for (int it = 0; it < (T_ / 8) / 8; ++it) {   // 64 iters x 8 t-rows
    const size_t o = (size_t)it * 8 * M_;
    v2f b0, b1;
    b0.x = p0[o];
    b0.y = p1[o];
    b1.x = p0[o + (size_t)4 * M_];
    b1.y = p1[o + (size_t)4 * M_];
#if HAS_WMMA_F32X4
    c0 = __builtin_amdgcn_wmma_f32_16x16x4_f32(false, ones, false, b0,
                                               (short)0, c0, false, false);
    c1 = __builtin_amdgcn_wmma_f32_16x16x4_f32(false, ones, false, b1,
                                               (short)0, c1, false, false);
#else
    c0[0] += b0.x + b0.y;            // per-half partial (rows rhalf, rhalf+2)
    c1[0] += b1.x + b1.y;
#endif
  }
  const float part = c0[0] + c1[0];

#if HAS_WMMA_F32X4
  // Both half-waves hold the full 4-row column sums -> use lanes 0..15 only.
  if (lane < 16) sp[w * 16 + lane] = part;
  __syncthreads();
  if (tid < 16) {
    float s = 0.f;
#pragma unroll
    for (int gg = 0; gg < 8; ++gg) s += sp[gg * 16 + tid];
    ksum[b * M_ + m0 + tid] = s * RSQRT10;     // pre-scale by 1/sqrt(D)
  }
#else
  // Scalar fallback: halves hold disjoint row pairs -> reduce 8 waves x 2 halves.
  sp[w * 32 + lane] = part;
  __syncthreads();
  if (tid < 16) {
    float s = 0.f;
#pragma unroll
    for (int gg = 0; gg < 8; ++gg)
      s += sp[gg * 32 + tid] + sp[gg * 32 + 16 + tid];
    ksum[b * M_ + m0 + tid] = s * RSQRT10;
  }
#endif
}

// ============================================================================
// Kernel 2: out[b,h,t] = sum_d softmax_d(tanh(q*ksum_scaled)) * v
// One block per (b, 16-row t tile). q/v tiles staged via async b128 -> LDS.
// LDS compute reads at word-stride 10 are bank-conflict-free in wave32.
// Output transposed through LDS so global stores are coalesced float4 along t.
// ============================================================================
__global__ __launch_bounds__(256) void attn_kernel(const float* __restrict__ q,
                                                   const float* __restrict__ v,
                                                   const float* __restrict__ ksum,
                                                   float* __restrict__ out) {
  __shared__ __align__(16) float s_q[TT * M_];     // 40 KB
  __shared__ __align__(16) float s_v[TT * M_];     // 40 KB
  __shared__             float s_ks[M_];           // 2.5 KB
  __shared__ __align__(16) float s_out[H_ * TT];   // 4 KB

  const int tid = threadIdx.x;
  const int b   = blockIdx.y;
  const int t0  = blockIdx.x * TT;
  const size_t base = ((size_t)b * T_ + (size_t)t0) * M_;

  // ---- stage q and v tiles with the async data mover (16B/lane/op) ----
  const char* gq = (const char*)(q + base);
  const char* gv = (const char*)(v + base);
  char* lq = (char*)s_q;
  char* lv = (char*)s_v;
#pragma unroll
  for (int i = 0; i < (TT * M_ * 4) / (16 * 256); ++i) {   // 10 ops each
    const int off = (i * 256 + tid) * 16;
    async_b128(gq + off, lq + off);
    async_b128(gv + off, lv + off);
  }

  // ksum row for this batch (already scaled by 1/sqrt(10))
  for (int i = tid; i < M_; i += 256) s_ks[i] = ksum[b * M_ + i];

  wait_async0();        // this wave's async copies complete
  __syncthreads();      // all waves' copies + ksum ds stores visible

  // ---- compute: thread -> (t_local, h); 4 items per thread ----
#pragma unroll
  for (int it = 0; it < (TT * H_) / 256; ++it) {
    const int item = it * 256 + tid;
    const int h    = item & (H_ - 1);
    const int tl   = item >> 6;
    const float* qr = &s_q[tl * M_ + h * D_];
    const float* vr = &s_v[tl * M_ + h * D_];
    const float* kr = &s_ks[h * D_];
    float se = 0.f, acc = 0.f;
#pragma unroll
    for (int d = 0; d < D_; ++d) {
      const float e = __expf(fast_tanh(qr[d] * kr[d]));  // tanh in [-1,1] -> safe
      se  += e;
      acc += e * vr[d];
    }
    s_out[h * TT + tl] = acc / se;
  }
  __syncthreads();

  // ---- coalesced float4 store: out layout [B,H,T] ----
  const int h   = tid >> 2;
  const int grp = tid & 3;
  const float4 val = *(const float4*)&s_out[h * TT + grp * 4];
  *(float4*)&out[((size_t)b * H_ + h) * T_ + (size_t)(t0 + grp * 4)] = val;
}

// ============================================================================
extern "C" void kernel_launch(void* const* d_in, const int* in_sizes, int n_in,
                              void* d_out, int out_size, void* d_ws, size_t ws_size,
                              hipStream_t stream) {
  const float* q = (const float*)d_in[0];   // [16,4096,640]
  const float* k = (const float*)d_in[1];   // [16,4096,640]
  const float* v = (const float*)d_in[2];   // [16,4096,640]
  // d_in[3] (W) and d_in[4] (b) are mathematically eliminated by the
  // softmax over a size-1 axis in the reference (s == 1).
  float* out  = (float*)d_out;              // [16,64,4096]
  float* ksum = (float*)d_ws;               // 16*640 floats = 40 KB scratch

  ksum_kernel<<<dim3(M_ / 16, B_), 256, 0, stream>>>(k, ksum);
  attn_kernel<<<dim3(T_ / TT, B_), 256, 0, stream>>>(q, v, ksum, out);
}